// TreeFilter2D_22789096473136
// MI455X (gfx1250) — compile-verified
//
#include <hip/hip_runtime.h>
#include <hip/hip_bf16.h>

// Problem constants (match the reference)
#define BATCH 8
#define CFEAT 512
#define CEMB  64
#define VN    4096      // H*W = 64*64
#define VG16  (VN / 16) // 16-node groups per batch
#define CPB   8         // feature channels per sweep block
#define NCH   (CPB + 1) // + ones channel
#define LVLS  4098      // level-offset stride per batch (depth < VN, +2 slack)

typedef float v2f __attribute__((ext_vector_type(2)));
typedef float v8f __attribute__((ext_vector_type(8)));

// ---------------------------------------------------------------------------
// Kernel 1: edge weights via V_WMMA_F32_16X16X4_F32.
// One wave32 handles 16 BFS-ordered nodes. D = src - parent (16x64 f32).
// Accumulate G = D * D^T with 16 WMMA K=4 steps; ew = exp(-diag(G)).
//
// f32 16x4 A layout (ISA 7.12.2): lanes 0-15 hold rows M=lane with
// VGPR0=K0,VGPR1=K1; lanes 16-31 hold the same rows with VGPR0=K2,VGPR1=K3.
// The 4x16 B layout for B = A^T has identical per-lane contents, so we pass
// the same v2f as both operands.
// ---------------------------------------------------------------------------
__global__ __launch_bounds__(256) void tf_edge_weights(
    const float* __restrict__ embed,        // (B, CE, V)
    const int*   __restrict__ sorted_index, // (B, V)
    const int*   __restrict__ sorted_parent,// (B, V)
    float*       __restrict__ ew_out)       // (B, V)
{
    const int lane = threadIdx.x & 31;
    const int wave = threadIdx.x >> 5;
    const int g    = blockIdx.x * 8 + wave;     // group id: 0 .. B*VG16-1
    const int b    = g / VG16;
    const int i0   = (g % VG16) * 16;

    const int r    = lane & 15;                 // matrix row this lane feeds
    const int koff = (lane >> 4) * 2;           // lanes 16-31 feed K=+2,+3
    const int node = i0 + r;

    const int j_src = sorted_index[b * VN + node];
    const int j_tar = sorted_index[b * VN + sorted_parent[b * VN + node]];
    const float* fmb = embed + (size_t)b * CEMB * VN;

    v8f acc = {0.f, 0.f, 0.f, 0.f, 0.f, 0.f, 0.f, 0.f};

#pragma unroll
    for (int kk = 0; kk < CEMB / 4; ++kk) {
        const int k = kk * 4 + koff;
        v2f d;
        d.x = fmb[(size_t)(k + 0) * VN + j_src] - fmb[(size_t)(k + 0) * VN + j_tar];
        d.y = fmb[(size_t)(k + 1) * VN + j_src] - fmb[(size_t)(k + 1) * VN + j_tar];
        // (neg_a, A, neg_b, B, c_mod, C, reuse_a, reuse_b)
        acc = __builtin_amdgcn_wmma_f32_16x16x4_f32(
            false, d, false, d, (short)0, acc, false, false);
    }

    // Diagonal extraction per C/D layout: (n,n) is lane n / VGPR n for n<8,
    // and lane n+16 / VGPR n-8 for n>=8.
    int sel = -1, row = 0;
    if (lane < 8)        { sel = lane;      row = lane;      }
    else if (lane >= 24) { sel = lane - 24; row = lane - 16; }
    if (sel >= 0) {
        float v = acc[0];
        switch (sel) {
            case 1: v = acc[1]; break;
            case 2: v = acc[2]; break;
            case 3: v = acc[3]; break;
            case 4: v = acc[4]; break;
            case 5: v = acc[5]; break;
            case 6: v = acc[6]; break;
            case 7: v = acc[7]; break;
            default: break;
        }
        ew_out[b * VN + i0 + row] = __expf(-v);
    }
}

// ---------------------------------------------------------------------------
// Kernel 2: per-depth level offsets. BFS order => node_depth is
// non-decreasing, so each depth occupies a contiguous range [lvl[d], lvl[d+1]).
// ---------------------------------------------------------------------------
__global__ __launch_bounds__(256) void tf_level_offsets(
    const int* __restrict__ node_depth, // (B, V)
    const int* __restrict__ max_depth,  // scalar
    int*       __restrict__ lvl)        // (B, LVLS)
{
    const int b = blockIdx.x;
    for (int i = threadIdx.x; i < VN; i += blockDim.x) {
        const int d = node_depth[b * VN + i];
        if (i == 0 || node_depth[b * VN + i - 1] != d)
            lvl[b * LVLS + d] = i;
    }
    if (threadIdx.x == 0)
        lvl[b * LVLS + *max_depth + 1] = VN;
}

// ---------------------------------------------------------------------------
// Kernel 3: resident tree sweep. One block = (batch, 8 feature channels) +
// the ones channel. All 9 channel vectors (9*16KB), ew, parent and index
// arrays live in 192KB of LDS (<=320KB/WGP on CDNA5). The 126-level up/down
// sweeps run inside the block with one s_barrier per level; up-sweep
// sibling->parent conflicts resolved with ds_add_f32 LDS atomics.
// ---------------------------------------------------------------------------
extern __shared__ unsigned char tf_smem[];

__global__ __launch_bounds__(1024) void tf_sweep(
    const float* __restrict__ feature,      // (B, C, V)
    const int*   __restrict__ sorted_index, // (B, V)
    const int*   __restrict__ sorted_parent,// (B, V)
    const float* __restrict__ ew_g,         // (B, V)
    const int*   __restrict__ lvl,          // (B, LVLS)
    const int*   __restrict__ max_depth,    // scalar
    float*       __restrict__ out)          // (B, C, V)
{
    float* A   = (float*)tf_smem;            // NCH * VN floats
    float* ews = A + (size_t)NCH * VN;       // VN floats
    int*   par = (int*)(ews + VN);           // VN ints
    int*   idx = par + VN;                   // VN ints

    const int b  = blockIdx.x >> 6;          // 64 channel-groups per batch
    const int c0 = (blockIdx.x & 63) * CPB;
    const int tid = threadIdx.x;
    const int nthr = blockDim.x;

    // Stage indices / weights, gather feature channels into BFS order.
    for (int i = tid; i < VN; i += nthr) {
        idx[i] = sorted_index[b * VN + i];
        par[i] = sorted_parent[b * VN + i];
        ews[i] = ew_g[b * VN + i];
    }
    __syncthreads();
#pragma unroll
    for (int ch = 0; ch < CPB; ++ch) {
        const float* src = feature + ((size_t)b * CFEAT + (c0 + ch)) * VN;
        for (int i = tid; i < VN; i += nthr)
            A[ch * VN + i] = src[idx[i]];
    }
    for (int i = tid; i < VN; i += nthr)
        A[CPB * VN + i] = 1.0f;              // ones channel
    __syncthreads();

    const int maxd = *max_depth;
    const int* lv = lvl + b * LVLS;

    // Up-sweep: leaves -> root.
    for (int d = maxd; d >= 1; --d) {
        const int s = lv[d];
        const int cnt = lv[d + 1] - s;
        if (cnt > 0) {
            for (int j = tid; j < cnt * NCH; j += nthr) {
                const int n  = s + (j % cnt);
                const int ch = j / cnt;
                atomicAdd(&A[ch * VN + par[n]], ews[n] * A[ch * VN + n]);
            }
        }
        __syncthreads();
    }

    // Down-sweep: root -> leaves (in place; parent level already final,
    // own value untouched since up-sweep => matches reference semantics).
    for (int d = 1; d <= maxd; ++d) {
        const int s = lv[d];
        const int cnt = lv[d + 1] - s;
        if (cnt > 0) {
            for (int j = tid; j < cnt * NCH; j += nthr) {
                const int n  = s + (j % cnt);
                const int ch = j / cnt;
                const float a = A[ch * VN + n];
                const float w = ews[n];
                A[ch * VN + n] = a + w * (A[ch * VN + par[n]] - w * a);
            }
        }
        __syncthreads();
    }

    // Normalize by ones channel and scatter back to image order.
#pragma unroll
    for (int ch = 0; ch < CPB; ++ch) {
        float* dst = out + ((size_t)b * CFEAT + (c0 + ch)) * VN;
        for (int i = tid; i < VN; i += nthr)
            dst[idx[i]] = A[ch * VN + i] / A[CPB * VN + i];
    }
}

// ---------------------------------------------------------------------------
extern "C" void kernel_launch(void* const* d_in, const int* in_sizes, int n_in,
                              void* d_out, int out_size, void* d_ws, size_t ws_size,
                              hipStream_t stream) {
    const float* feature = (const float*)d_in[0];
    const float* embed   = (const float*)d_in[1];
    const int*   sidx    = (const int*)d_in[2];
    const int*   spar    = (const int*)d_in[3];
    const int*   sdep    = (const int*)d_in[4];
    const int*   maxd    = (const int*)d_in[5];
    float*       out     = (float*)d_out;

    float* ew  = (float*)d_ws;                                   // B*V floats
    int*   lvl = (int*)((char*)d_ws + (size_t)BATCH * VN * 4);   // B*LVLS ints

    // 1) Edge weights (WMMA Gram diagonal). B*V/16 waves, 8 waves/block.
    tf_edge_weights<<<(BATCH * VG16) / 8, 256, 0, stream>>>(embed, sidx, spar, ew);

    // 2) Level offsets.
    tf_level_offsets<<<BATCH, 256, 0, stream>>>(sdep, maxd, lvl);

    // 3) Resident LDS tree sweep: 192KB dynamic LDS per block.
    const size_t smem = ((size_t)NCH * VN + VN) * sizeof(float) + 2 * VN * sizeof(int);
    tf_sweep<<<BATCH * (CFEAT / CPB), 1024, smem, stream>>>(
        feature, sidx, spar, ew, lvl, maxd, out);
}